// Global_85143431676036
// MI455X (gfx1250) — compile-verified
//
#include <hip/hip_runtime.h>

#define B    8
#define C    256
#define H    48
#define W    48
#define HW   (H * W)          // 2304
#define TOPK 256
#define SORTN 4096            // next pow2 >= HW

typedef __attribute__((ext_vector_type(2))) float v2f;
typedef __attribute__((ext_vector_type(8))) float v8f;

// ---------------------------------------------------------------------------
// Kernel 1: 2x2 block max-pool of mask -> per-row scale m[b,p]
// ---------------------------------------------------------------------------
__global__ __launch_bounds__(256) void pool_mask_kernel(
    const float* __restrict__ mask, float* __restrict__ mvec) {
  int id = blockIdx.x * 256 + threadIdx.x;
  if (id >= B * HW) return;
  int b = id / HW, p = id % HW;
  int h = p / W, w = p % W;
  const float* mb = mask + (size_t)b * (2 * H) * (2 * W);
  int r = (2 * h) * (2 * W) + 2 * w;
  float m0 = mb[r], m1 = mb[r + 1], m2 = mb[r + 2 * W], m3 = mb[r + 2 * W + 1];
  mvec[id] = fmaxf(fmaxf(m0, m1), fmaxf(m2, m3));
}

// ---------------------------------------------------------------------------
// Kernel 2: reciprocal L2 norms over channels for key and curr
// ---------------------------------------------------------------------------
__global__ __launch_bounds__(256) void rnorm_kernel(
    const float* __restrict__ key, const float* __restrict__ curr,
    float* __restrict__ rk, float* __restrict__ rc) {
  int id = blockIdx.x * 256 + threadIdx.x;
  if (id >= B * HW) return;
  int b = id / HW, s = id % HW;
  const float* kb = key  + (size_t)b * C * HW + s;
  const float* cb = curr + (size_t)b * C * HW + s;
  float sk = 0.f, sc = 0.f;
  for (int c = 0; c < C; ++c) {
    float a = kb[(size_t)c * HW]; sk += a * a;
    float d = cb[(size_t)c * HW]; sc += d * d;
  }
  rk[id] = 1.0f / fmaxf(sqrtf(sk), 1e-12f);
  rc[id] = 1.0f / fmaxf(sqrtf(sc), 1e-12f);
}

// ---------------------------------------------------------------------------
// Kernel 3: normalized curr, same layout Cn[b][c][s] (B-matrix of GEMM)
// ---------------------------------------------------------------------------
__global__ __launch_bounds__(256) void norm_curr_kernel(
    const float* __restrict__ curr, const float* __restrict__ rc,
    float* __restrict__ Cn) {
  int idx = blockIdx.x * 256 + threadIdx.x;
  if (idx >= B * C * HW) return;
  int s = idx % HW;
  int b = idx / (C * HW);
  Cn[idx] = curr[idx] * rc[b * HW + s];
}

// ---------------------------------------------------------------------------
// Kernel 4: normalized + transposed key, Knt[b][p][c] (A-matrix of GEMM)
// ---------------------------------------------------------------------------
__global__ __launch_bounds__(256) void tkey_kernel(
    const float* __restrict__ key, const float* __restrict__ rk,
    float* __restrict__ Knt) {
  __shared__ float tile[32][33];
  int b  = blockIdx.z;
  int c0 = blockIdx.y * 32;
  int s0 = blockIdx.x * 32;
  int tx = threadIdx.x, ty = threadIdx.y;
#pragma unroll
  for (int r0 = 0; r0 < 4; ++r0) {
    int cl = ty + r0 * 8;
    tile[cl][tx] = key[((size_t)(b * C + c0 + cl)) * HW + s0 + tx];
  }
  __syncthreads();
#pragma unroll
  for (int r0 = 0; r0 < 4; ++r0) {
    int sl = ty + r0 * 8;
    int s  = s0 + sl;
    Knt[((size_t)(b * HW + s)) * C + c0 + tx] = tile[tx][sl] * rk[b * HW + s];
  }
}

// ---------------------------------------------------------------------------
// Kernel 5: per-batch correlation GEMM with V_WMMA_F32_16X16X4_F32.
// One wave32 per 64x32 output tile: 4x2 WMMA tiles (8 WMMA per K-step of 4).
// Widened M amortizes the strided B-fragment loads (4 b64 + 4 b32 -> 8 WMMA).
// A (16x4 f32): lane<16 holds A[M=l][K=0..1], lane>=16 holds A[M=l][K=2..3].
// B (4x16 f32): vgpr0 = rows K=0 (lanes 0-15) / K=2 (lanes 16-31), vgpr1 = K=1/K=3.
// ---------------------------------------------------------------------------
#define TM 4   // 16-row tiles per wave (M = 64)
#define TN 2   // 16-col tiles per wave (N = 32)

__global__ __launch_bounds__(256) void corr_gemm_kernel(
    const float* __restrict__ Knt, const float* __restrict__ Cn,
    float* __restrict__ corr) {
  const int MT = HW / (16 * TM);  // 36 M-tiles per batch
  const int NT = HW / (16 * TN);  // 72 N-tiles per batch
  int wid  = blockIdx.x * 8 + (threadIdx.x >> 5);
  int lane = threadIdx.x & 31;
  int l  = lane & 15;
  int hi = lane >> 4;

  int b  = wid / (MT * NT);
  int t  = wid % (MT * NT);
  int p0 = (t / NT) * (16 * TM);
  int q0 = (t % NT) * (16 * TN);

  const float* Kb = Knt + (size_t)b * HW * C;
  const float* Cb = Cn  + (size_t)b * C * HW;

  v8f acc[TM][TN];
#pragma unroll
  for (int mi = 0; mi < TM; ++mi)
#pragma unroll
    for (int ni = 0; ni < TN; ++ni) acc[mi][ni] = (v8f){};

  const float* arow[TM];
#pragma unroll
  for (int mi = 0; mi < TM; ++mi)
    arow[mi] = Kb + (size_t)(p0 + 16 * mi + l) * C + 2 * hi;

  for (int k = 0; k < C; k += 4) {
    v2f a[TM];
#pragma unroll
    for (int mi = 0; mi < TM; ++mi)
      a[mi] = *(const v2f*)(arow[mi] + k);   // 8B aligned: k%4==0, 2*hi even

    const float* crow = Cb + (size_t)(k + 2 * hi) * HW + q0;
    v2f bb[TN];
#pragma unroll
    for (int ni = 0; ni < TN; ++ni) {
      bb[ni].x = crow[16 * ni + l];
      bb[ni].y = crow[HW + 16 * ni + l];
    }

#pragma unroll
    for (int mi = 0; mi < TM; ++mi)
#pragma unroll
      for (int ni = 0; ni < TN; ++ni)
        acc[mi][ni] = __builtin_amdgcn_wmma_f32_16x16x4_f32(
            false, a[mi], false, bb[ni], (short)0, acc[mi][ni], false, false);
  }

  // C/D layout: vgpr r -> row M = r + 8*hi, column N = l
  size_t base = ((size_t)b * HW + p0) * HW + q0;
#pragma unroll
  for (int mi = 0; mi < TM; ++mi) {
#pragma unroll
    for (int r = 0; r < 8; ++r) {
      size_t row = base + (size_t)(16 * mi + r + 8 * hi) * HW;
#pragma unroll
      for (int ni = 0; ni < TN; ++ni)
        corr[row + 16 * ni + l] = acc[mi][ni][r];
    }
  }
}

// ---------------------------------------------------------------------------
// Kernel 6: per-column sorted top-256 (fwd & bwd mask scaling) via bitonic
// sort of 4096 padded values in LDS. One block per (b,q).
// ---------------------------------------------------------------------------
__device__ __forceinline__ void bitonic_sort_desc(float* buf, unsigned tid) {
  for (unsigned k = 2; k <= SORTN; k <<= 1) {
    for (unsigned j = k >> 1; j > 0; j >>= 1) {
      __syncthreads();
#pragma unroll
      for (unsigned w = 0; w < SORTN / 256; ++w) {
        unsigned i   = tid + (w << 8);
        unsigned ixj = i ^ j;
        if (ixj > i) {
          float a = buf[i];
          float c = buf[ixj];
          bool desc = ((i & k) == 0);
          if (desc ? (a < c) : (a > c)) { buf[i] = c; buf[ixj] = a; }
        }
      }
    }
  }
  __syncthreads();
}

__global__ __launch_bounds__(256) void topk_kernel(
    const float* __restrict__ corr, const float* __restrict__ mvec,
    float* __restrict__ out) {
  __shared__ float raw[HW];
  __shared__ float mloc[HW];
  __shared__ float buf[SORTN];

  unsigned tid = threadIdx.x;
  int b = blockIdx.x / HW;
  int q = blockIdx.x % HW;

  const float* col = corr + (size_t)b * HW * HW + q;
#pragma unroll
  for (int w = 0; w < HW / 256; ++w) {
    int p = tid + w * 256;
    raw[p]  = col[(size_t)p * HW];
    mloc[p] = mvec[b * HW + p];
  }
  __syncthreads();

  // forward branch: v * m
#pragma unroll
  for (unsigned w = 0; w < SORTN / 256; ++w) {
    unsigned i = tid + (w << 8);
    buf[i] = (i < HW) ? raw[i] * mloc[i] : -__builtin_inff();
  }
  bitonic_sort_desc(buf, tid);
  out[((size_t)(b * TOPK) + tid) * HW + q] = buf[tid];
  __syncthreads();

  // backward branch: v * (1 - m)
#pragma unroll
  for (unsigned w = 0; w < SORTN / 256; ++w) {
    unsigned i = tid + (w << 8);
    buf[i] = (i < HW) ? raw[i] * (1.0f - mloc[i]) : -__builtin_inff();
  }
  bitonic_sort_desc(buf, tid);
  out[(size_t)B * TOPK * HW + ((size_t)(b * TOPK) + tid) * HW + q] = buf[tid];
}

// ---------------------------------------------------------------------------
// Host launcher
// ---------------------------------------------------------------------------
extern "C" void kernel_launch(void* const* d_in, const int* in_sizes, int n_in,
                              void* d_out, int out_size, void* d_ws, size_t ws_size,
                              hipStream_t stream) {
  const float* key  = (const float*)d_in[0];
  const float* curr = (const float*)d_in[1];
  const float* mask = (const float*)d_in[2];
  float* out = (float*)d_out;

  char* ws = (char*)d_ws;
  size_t o = 0;
  float* Knt  = (float*)(ws + o); o += (size_t)B * HW * C * sizeof(float);
  float* Cn   = (float*)(ws + o); o += (size_t)B * C * HW * sizeof(float);
  float* rk   = (float*)(ws + o); o += (size_t)B * HW * sizeof(float);
  float* rc   = (float*)(ws + o); o += (size_t)B * HW * sizeof(float);
  float* mvec = (float*)(ws + o); o += (size_t)B * HW * sizeof(float);
  float* corr = (float*)(ws + o); o += (size_t)B * HW * HW * sizeof(float);
  (void)ws_size; (void)in_sizes; (void)n_in; (void)out_size;

  pool_mask_kernel<<<(B * HW + 255) / 256, 256, 0, stream>>>(mask, mvec);
  rnorm_kernel<<<(B * HW + 255) / 256, 256, 0, stream>>>(key, curr, rk, rc);
  norm_curr_kernel<<<(B * C * HW + 255) / 256, 256, 0, stream>>>(curr, rc, Cn);
  tkey_kernel<<<dim3(HW / 32, C / 32, B), dim3(32, 8), 0, stream>>>(key, rk, Knt);
  // waves = B * MT * NT = 8 * 36 * 72 = 20736 ; 8 waves per 256-thread block
  corr_gemm_kernel<<<(B * (HW / 64) * (HW / 32)) / 8, 256, 0, stream>>>(Knt, Cn, corr);
  topk_kernel<<<B * HW, 256, 0, stream>>>(corr, mvec, out);
}